// VectorKNN_87986700026093
// MI455X (gfx1250) — compile-verified
//
#include <hip/hip_runtime.h>
#include <math.h>

typedef __attribute__((ext_vector_type(2))) float v2f;
typedef __attribute__((ext_vector_type(8))) float v8f;

#define D_FEAT 256
#define WAVES 4
#define TPB (WAVES * 32)            // 128 threads
#define TILES_PER_WAVE 8
#define ROWS_PER_BLOCK (WAVES * 16 * TILES_PER_WAVE)  // 512
#define PH_COLS 64                  // K columns staged per phase
#define RS 68                       // padded LDS row stride (dwords): 68 % 64 == 4 -> conflict-free
#define NPH (TILES_PER_WAVE * 4)    // 32 phases per wave (4 K-phases per 16-row tile)
#define NEG_BIG (-3.402823466e38f)

// ---- small fixed top-5 insertion (fully unrolled -> registers) ----
__device__ __forceinline__ void ins5(float v, int idx, float tv[5], int ti[5]) {
#pragma unroll
    for (int j = 0; j < 5; ++j) {
        if (v > tv[j]) {
#pragma unroll
            for (int m = 4; m > 0; --m) {
                if (m > j) { tv[m] = tv[m - 1]; ti[m] = ti[m - 1]; }
            }
            tv[j] = v; ti[j] = idx;
            break;
        }
    }
}

// ---- CDNA5 async global->LDS copy (ASYNCcnt-tracked, no VGPR round-trip) ----
__device__ __forceinline__ void async_ld_b128(unsigned lds_addr, const float* gaddr) {
    asm volatile("global_load_async_to_lds_b128 %0, %1, off"
                 :: "v"(lds_addr), "v"(gaddr) : "memory");
}
__device__ __forceinline__ void wait_async_le8() {
    asm volatile("s_wait_asynccnt 0x8" ::: "memory");
}
__device__ __forceinline__ void wait_async_0() {
    asm volatile("s_wait_asynccnt 0x0" ::: "memory");
}
__device__ __forceinline__ void wait_ds_0() {
    asm volatile("s_wait_dscnt 0x0" ::: "memory");
}

// Issue one phase: 16 rows x 64 cols -> LDS buffer, 8 async b128 (2 rows per instr).
__device__ __forceinline__ void issue_phase(const float* __restrict__ bank,
                                            long long rowBase, int colBase,
                                            unsigned ldsBase, int laneHi, int colLane,
                                            int nrows)
{
    // buffer reuse guard: any ds_loads still draining from 2 phases ago must land first
    wait_ds_0();
    unsigned lds = ldsBase + (unsigned)((laneHi * RS + colLane) * 4);
#pragma unroll
    for (int r2 = 0; r2 < 8; ++r2) {
        long long row = rowBase + r2 * 2 + laneHi;
        if (row > (long long)nrows - 1) row = nrows - 1;   // clamp: OOB sims masked later
        const float* g = bank + (size_t)row * D_FEAT + colBase + colLane;
        async_ld_b128(lds, g);
        lds += (unsigned)(2 * RS * 4);
    }
}

// ============================================================
// Kernel 1: WMMA f32 GEMV (sims = q . bank_row) + per-block top-5
// ============================================================
__global__ __launch_bounds__(TPB) void knn_gemv(
        const float* __restrict__ q,
        const float* __restrict__ bank,
        float* __restrict__ cand_val,
        int*   __restrict__ cand_idx,
        int nrows)
{
    __shared__ float qlds[D_FEAT];                    // 1 KB
    __shared__ float tile[WAVES][2][16 * RS];         // 34,816 B (double-buffered per wave)
    __shared__ float sims[ROWS_PER_BLOCK];            // 2 KB

    const int tid     = threadIdx.x;
    const int wave    = tid >> 5;
    const int lane    = tid & 31;
    const int n       = lane & 15;                    // B-matrix column / tile row
    const int laneHi  = lane >> 4;
    const int sel     = laneHi * 2;                   // K sub-offset per lane half
    const int colLane = (lane & 15) * 4;              // staging column per lane

    // stage query once per block
    qlds[tid]       = q[tid];
    qlds[tid + TPB] = q[tid + TPB];
    __syncthreads();

    const long long blockBase = (long long)blockIdx.x * ROWS_PER_BLOCK;

    const unsigned ldsBuf0 = (unsigned)(uintptr_t)&tile[wave][0][0];
    const unsigned ldsBuf1 = (unsigned)(uintptr_t)&tile[wave][1][0];

    // ---- software pipeline over NPH phases (4 K-phases per 16-row tile) ----
    // prologue: phase 0
    {
        const long long rowBase0 = blockBase + (long long)(0 * WAVES + wave) * 16;
        issue_phase(bank, rowBase0, 0, ldsBuf0, laneHi, colLane, nrows);
    }

    v8f acc = {0.f, 0.f, 0.f, 0.f, 0.f, 0.f, 0.f, 0.f};

    for (int i = 0; i < NPH; ++i) {
        const int t       = i >> 2;
        const int ph      = i & 3;
        const int tileIdx = t * WAVES + wave;
        const long long rowBase = blockBase + (long long)tileIdx * 16;

        // issue next phase into the other buffer while this one computes
        if (i + 1 < NPH) {
            const int in       = i + 1;
            const int tn       = in >> 2;
            const long long rb = blockBase + (long long)(tn * WAVES + wave) * 16;
            issue_phase(bank, rb, (in & 3) * PH_COLS,
                        ((in & 1) ? ldsBuf1 : ldsBuf0), laneHi, colLane, nrows);
            wait_async_le8();       // in-order completion: phase i has landed
        } else {
            wait_async_0();
        }

        // ---- 16 WMMA steps over this 64-col K slab ----
        const float* tbuf = &tile[wave][i & 1][0];
        const int colBase = ph * PH_COLS;
#pragma unroll
        for (int k0 = 0; k0 < PH_COLS; k0 += 4) {
            // A (16x4 f32): query chunk replicated over M
            v2f a = *(const v2f*)(&qlds[colBase + k0 + sel]);
            // B (4x16 f32): 16 bank rows as columns (conflict-free, RS=68)
            v2f b = *(const v2f*)(&tbuf[n * RS + k0 + sel]);
            acc = __builtin_amdgcn_wmma_f32_16x16x4_f32(
                    false, a, false, b, (short)0, acc, false, false);
        }

        // end of a 16-row tile: publish sims, reset accumulator
        if (ph == 3) {
            if (lane < 16) {
                long long row = rowBase + lane;
                sims[tileIdx * 16 + lane] = (row < nrows) ? acc[0] : NEG_BIG;
            }
            acc = (v8f){0.f, 0.f, 0.f, 0.f, 0.f, 0.f, 0.f, 0.f};
        }
    }
    __syncthreads();

    // ---- block-level top-5 (wave 0 lanes -> LDS -> thread 0) ----
    float* lv = &tile[0][0][0];
    int*   li = (int*)&tile[1][0][0];
    if (wave == 0) {
        float tv[5]; int ti[5];
#pragma unroll
        for (int j = 0; j < 5; ++j) { tv[j] = NEG_BIG; ti[j] = 0; }
        for (int i = lane; i < ROWS_PER_BLOCK; i += 32)
            ins5(sims[i], i, tv, ti);
#pragma unroll
        for (int j = 0; j < 5; ++j) { lv[lane * 5 + j] = tv[j]; li[lane * 5 + j] = ti[j]; }
    }
    // wave-0 LDS ops are in-order; thread 0 merges
    if (tid == 0) {
        float fv[5]; int fi[5];
#pragma unroll
        for (int j = 0; j < 5; ++j) { fv[j] = NEG_BIG; fi[j] = 0; }
        for (int i = 0; i < 32 * 5; ++i)
            ins5(lv[i], li[i], fv, fi);
#pragma unroll
        for (int j = 0; j < 5; ++j) {
            cand_val[blockIdx.x * 5 + j] = fv[j];
            cand_idx[blockIdx.x * 5 + j] = (int)(blockBase + fi[j]);
        }
    }
}

// ============================================================
// Kernel 2: single-block final merge + query-norm scaling
// ============================================================
__global__ __launch_bounds__(256) void knn_final(
        const float* __restrict__ q,
        const float* __restrict__ cv,
        const int*   __restrict__ ci,
        int ncand,
        float* __restrict__ out)
{
    __shared__ float red[256];
    __shared__ float lv[256 * 5];
    __shared__ int   li[256 * 5];

    const int t = threadIdx.x;

    // ||q||^2 reduction (D_FEAT == 256 == blockDim)
    float x = q[t];
    red[t] = x * x;
    __syncthreads();
    for (int s = 128; s > 0; s >>= 1) {
        if (t < s) red[t] += red[t + s];
        __syncthreads();
    }

    float tv[5]; int ti[5];
#pragma unroll
    for (int j = 0; j < 5; ++j) { tv[j] = NEG_BIG; ti[j] = 0; }
    for (int i = t; i < ncand; i += 256)
        ins5(cv[i], ci[i], tv, ti);
#pragma unroll
    for (int j = 0; j < 5; ++j) { lv[t * 5 + j] = tv[j]; li[t * 5 + j] = ti[j]; }
    __syncthreads();

    if (t == 0) {
        float fv[5]; int fi[5];
#pragma unroll
        for (int j = 0; j < 5; ++j) { fv[j] = NEG_BIG; fi[j] = 0; }
        for (int i = 0; i < 256 * 5; ++i)
            ins5(lv[i], li[i], fv, fi);

        float norm = sqrtf(red[0]);
        float inv  = 1.0f / fmaxf(norm, 1e-12f);
        int* oi = (int*)out;                 // tuple: (top_idx[5], top_vals[5])
        for (int j = 0; j < 5; ++j) {
            oi[j]      = fi[j];
            out[5 + j] = fv[j] * inv;
        }
    }
}

// ============================================================
extern "C" void kernel_launch(void* const* d_in, const int* in_sizes, int n_in,
                              void* d_out, int out_size, void* d_ws, size_t ws_size,
                              hipStream_t stream) {
    const float* q    = (const float*)d_in[0];
    const float* bank = (const float*)d_in[1];
    const int nrows   = in_sizes[1] / D_FEAT;

    const int nb = (nrows + ROWS_PER_BLOCK - 1) / ROWS_PER_BLOCK;

    float* cand_val = (float*)d_ws;
    int*   cand_idx = (int*)((char*)d_ws + (size_t)nb * 5 * sizeof(float));

    knn_gemv<<<nb, TPB, 0, stream>>>(q, bank, cand_val, cand_idx, nrows);
    knn_final<<<1, 256, 0, stream>>>(q, cand_val, cand_idx, nb * 5, (float*)d_out);
}